// NLSearch_58050777973397
// MI455X (gfx1250) — compile-verified
//
#include <hip/hip_runtime.h>

typedef __attribute__((ext_vector_type(2))) float v2f;
typedef __attribute__((ext_vector_type(8))) float v8f;

#define PSZ      7
#define HH       64
#define OH       58            // 64 - 7 + 1
#define LL       3364          // 58*58
#define CIN      32
#define DIMK     1568          // 32*49
#define KC       32            // k-chunk width
#define NCHUNK   49            // 1568 / 32
#define TM       128
#define TN       128
#define LDSS     33            // padded LDS stride (odd -> conflict-free)

__global__ __launch_bounds__(256)
void nls_attn_wmma(const float* __restrict__ x, float* __restrict__ out)
{
    // attn = q q^T is symmetric per image: only compute tiles with by <= bx,
    // mirror-store the strictly off-diagonal ones.  (block-uniform early exit)
    if (blockIdx.y > blockIdx.x) return;

    __shared__ float sA[TM][LDSS];
    __shared__ float sB[TN][LDSS];

    const int n     = blockIdx.z;            // image 0..7
    const int mBase = blockIdx.y * TM;       // output row tile
    const int nBase = blockIdx.x * TN;       // output col tile
    const bool offDiag = (blockIdx.y != blockIdx.x);
    const float* __restrict__ xs = x + (size_t)n * CIN * HH * HH;

    const int t    = threadIdx.x;
    const int lane = t & 31;                 // wave32
    const int w    = t >> 5;                 // wave id 0..7
    const int wm   = w & 3;                  // which 32-row strip of 128
    const int wn   = w >> 2;                 // which 64-col half of 128

    // WMMA f32 16x16x4 fragment indices (ISA 7.12.2, 32-bit A 16x4 layout)
    const int fm  = lane & 15;               // M (for A) / N (for B)
    const int fkb = (lane >> 4) << 1;        // K base: 0 (lanes 0-15) or 2 (lanes 16-31)
    const int hi  = lane >> 4;

    v8f acc[2][4];
    #pragma unroll
    for (int mi = 0; mi < 2; ++mi)
        #pragma unroll
        for (int ni = 0; ni < 4; ++ni)
            acc[mi][ni] = (v8f){};

    // ---- fused im2col: thread owns fixed k-lane kk; rows i = e*8 + rowOff ----
    const int kk     = t & 31;
    const int rowOff = t >> 5;

    // Hoist the patch-position decode (independent of k-chunk): spatial offset
    // of patch l's origin inside one channel plane, or -1 if row is OOB.
    int offA[16], offB[16];
    #pragma unroll
    for (int e = 0; e < 16; ++e) {
        const int i = e * 8 + rowOff;
        int l = mBase + i;
        if (l < LL) { const int y = l / OH; offA[e] = y * HH + (l - y * OH); }
        else        { offA[e] = -1; }
        l = nBase + i;
        if (l < LL) { const int y = l / OH; offB[e] = y * HH + (l - y * OH); }
        else        { offB[e] = -1; }
    }

    float pa[16], pb[16];

    // gather one 32-wide k-chunk into registers
    auto gather = [&](int ck) {
        const int d   = ck * KC + kk;        // feature index
        const int ch  = d / 49;
        const int rem = d - ch * 49;
        const int dy  = rem / PSZ;
        const int dx  = rem - dy * PSZ;
        const int srcBase = ch * (HH * HH) + dy * HH + dx;
        #pragma unroll
        for (int e = 0; e < 16; ++e) {
            pa[e] = (offA[e] >= 0) ? xs[srcBase + offA[e]] : 0.0f;
            pb[e] = (offB[e] >= 0) ? xs[srcBase + offB[e]] : 0.0f;
        }
    };

    gather(0);

    for (int ck = 0; ck < NCHUNK; ++ck) {
        __syncthreads();                     // prior compute done reading LDS
        #pragma unroll
        for (int e = 0; e < 16; ++e) {
            const int i = e * 8 + rowOff;
            sA[i][kk] = pa[e];
            sB[i][kk] = pb[e];
        }
        __syncthreads();                     // tile published

        if (ck + 1 < NCHUNK) gather(ck + 1); // next chunk's loads in flight

        // 8 k-steps of 4; per step: 2 A-frags x 4 B-frags -> 8 WMMAs
        #pragma unroll
        for (int ks = 0; ks < KC / 4; ++ks) {
            const int kb = ks * 4 + fkb;
            v2f a[2], b[4];
            #pragma unroll
            for (int mi = 0; mi < 2; ++mi) {
                a[mi].x = sA[32 * wm + 16 * mi + fm][kb];
                a[mi].y = sA[32 * wm + 16 * mi + fm][kb + 1];
            }
            #pragma unroll
            for (int ni = 0; ni < 4; ++ni) {
                b[ni].x = sB[64 * wn + 16 * ni + fm][kb];
                b[ni].y = sB[64 * wn + 16 * ni + fm][kb + 1];
            }
            #pragma unroll
            for (int mi = 0; mi < 2; ++mi)
                #pragma unroll
                for (int ni = 0; ni < 4; ++ni)
                    acc[mi][ni] = __builtin_amdgcn_wmma_f32_16x16x4_f32(
                        false, a[mi], false, b[ni], (short)0, acc[mi][ni],
                        false, false);
        }
    }

    // ---- store C/D: VGPR r -> M = r + 8*hi, N = lane&15 ----
    // Off-diagonal tiles additionally store the mirrored tile (attn symmetric);
    // per lane the transposed writes hit 8 consecutive rows -> 32B chunks.
    float* __restrict__ outn = out + (size_t)n * LL * LL;
    #pragma unroll
    for (int mi = 0; mi < 2; ++mi) {
        #pragma unroll
        for (int ni = 0; ni < 4; ++ni) {
            const int col = nBase + 64 * wn + 16 * ni + fm;
            #pragma unroll
            for (int r = 0; r < 8; ++r) {
                const int row = mBase + 32 * wm + 16 * mi + r + 8 * hi;
                if (row < LL && col < LL) {
                    const float v = acc[mi][ni][r];
                    __builtin_nontemporal_store(v, &outn[(size_t)row * LL + col]);
                    if (offDiag)
                        __builtin_nontemporal_store(v, &outn[(size_t)col * LL + row]);
                }
            }
        }
    }
}

__global__ void nls_inds_zero(float* __restrict__ out)
{
    out[(size_t)8 * LL * LL] = 0.0f;         // trailing `inds` scalar
}

extern "C" void kernel_launch(void* const* d_in, const int* in_sizes, int n_in,
                              void* d_out, int out_size, void* d_ws, size_t ws_size,
                              hipStream_t stream)
{
    (void)in_sizes; (void)n_in; (void)out_size; (void)d_ws; (void)ws_size;
    const float* x = (const float*)d_in[0];
    float* out = (float*)d_out;

    dim3 grid((LL + TN - 1) / TN, (LL + TM - 1) / TM, 8);   // 27 x 27 x 8
    nls_attn_wmma<<<grid, 256, 0, stream>>>(x, out);
    nls_inds_zero<<<1, 1, 0, stream>>>(out);
}